// ForgetMult_19000935318057
// MI455X (gfx1250) — compile-verified
//
#include <hip/hip_runtime.h>
#include <stdint.h>

// Problem geometry (fixed by the reference).
#define SEQ    4096
#define BATCH  16
#define HID    1024
#define CH     (BATCH * HID)      // 16384 independent channels, contiguous axis
#define NCHUNK 32                 // chunks along SEQ (parallelism multiplier)
#define TCHUNK (SEQ / NCHUNK)     // 128 timesteps per chunk
#define BLOCK  256
#define CPT    4                  // channels per thread (float4 / B128)
#define CHW    (BLOCK * CPT)      // 1024 channels per workgroup
#define NCB    (CH / CHW)         // 16 channel blocks
#define DSLOT  8                  // LDS circular-buffer slots (power of two)
#define KAHEAD 6                  // timesteps kept in flight via ASYNCcnt

typedef __attribute__((ext_vector_type(4))) float v4f;   // native vector for NT store

// ---- CDNA5 async global->LDS copy path (ASYNCcnt-tracked, VGPR-free prefetch) ----
__device__ __forceinline__ void async_ld16(uint32_t lds_addr, const float* gaddr) {
    // GV mode: 64-bit global address in VGPR pair, SADDR=off; VDST = per-lane LDS byte addr.
    asm volatile("global_load_async_to_lds_b128 %0, %1, off"
                 :: "v"(lds_addr), "v"(gaddr) : "memory");
}
__device__ __forceinline__ void wait_async_steady() {
    // <= 2*(KAHEAD-1) outstanding loads -> oldest (f,x) row pair has landed in LDS.
    asm volatile("s_wait_asynccnt 10" ::: "memory");
}
__device__ __forceinline__ void wait_async_all() {
    asm volatile("s_wait_asynccnt 0" ::: "memory");
}
__device__ __forceinline__ void wait_ds_all() {
    // Retire our ds_loads before the async engine overwrites that LDS slot (WAR guard).
    asm volatile("s_wait_dscnt 0" ::: "memory");
}

// ---------------- Phase 1: per-chunk scan summaries A = prod(1-f), B = local h ----------------
__global__ __launch_bounds__(BLOCK) void fm_phase1(const float* __restrict__ f,
                                                   const float* __restrict__ x,
                                                   float* __restrict__ Aout,
                                                   float* __restrict__ Bout) {
    __shared__ float4 bF[DSLOT][BLOCK];   // 32 KB
    __shared__ float4 bX[DSLOT][BLOCK];   // 32 KB
    const int tid    = threadIdx.x;
    const int chunk  = blockIdx.y;
    const int chBase = blockIdx.x * CHW + tid * CPT;

    const float* fp = f + (size_t)chunk * TCHUNK * CH + chBase;
    const float* xp = x + (size_t)chunk * TCHUNK * CH + chBase;

    const uint32_t lF = (uint32_t)(uintptr_t)&bF[0][tid];
    const uint32_t lX = (uint32_t)(uintptr_t)&bX[0][tid];
    const uint32_t slotBytes = (uint32_t)(sizeof(float4) * BLOCK);  // 4096 B per row slot

    // Prologue: put KAHEAD (f,x) row pairs in flight. Each wave loads only its own lanes'
    // 16B slices, and later consumes only those slices -> no cross-wave sync needed.
    #pragma unroll
    for (int p = 0; p < KAHEAD; ++p) {
        async_ld16(lF + (uint32_t)p * slotBytes, fp + (size_t)p * CH);
        async_ld16(lX + (uint32_t)p * slotBytes, xp + (size_t)p * CH);
    }

    float ax = 1.f, ay = 1.f, az = 1.f, aw = 1.f;   // A accumulators
    float bx = 0.f, by = 0.f, bz = 0.f, bw = 0.f;   // B accumulators

    for (int i = 0; i < TCHUNK; ++i) {
        if (i < TCHUNK - KAHEAD)       wait_async_steady();
        else if (i == TCHUNK - KAHEAD) wait_async_all();   // tail: everything already issued

        const int slot = i & (DSLOT - 1);
        const float4 f4 = bF[slot][tid];
        const float4 x4 = bX[slot][tid];

        // B <- f*x + (1-f)*B == B + f*(x - B);  A <- (1-f)*A == A - f*A
        bx += f4.x * (x4.x - bx);  ax -= f4.x * ax;
        by += f4.y * (x4.y - by);  ay -= f4.y * ay;
        bz += f4.z * (x4.z - bz);  az -= f4.z * az;
        bw += f4.w * (x4.w - bw);  aw -= f4.w * aw;

        const int nr = i + KAHEAD;
        if (nr < TCHUNK) {
            wait_ds_all();
            const int ns = nr & (DSLOT - 1);
            async_ld16(lF + (uint32_t)ns * slotBytes, fp + (size_t)nr * CH);
            async_ld16(lX + (uint32_t)ns * slotBytes, xp + (size_t)nr * CH);
        }
    }

    const size_t o = (size_t)chunk * CH + chBase;
    float4 A4; A4.x = ax; A4.y = ay; A4.z = az; A4.w = aw;
    float4 B4; B4.x = bx; B4.y = by; B4.z = bz; B4.w = bw;
    *(float4*)(Aout + o) = A4;   // re-read by phase 2 shortly -> keep regular-temporal
    *(float4*)(Bout + o) = B4;
}

// ---------------- Phase 2: serial scan over 32 chunk summaries per channel ----------------
// Rewrites Barr[c][ch] in place with the *starting* h for chunk c (exclusive scan).
__global__ __launch_bounds__(BLOCK) void fm_phase2(float* __restrict__ Aarr,
                                                   float* __restrict__ Barr,
                                                   const float* __restrict__ h0) {
    const int ch = blockIdx.x * BLOCK + threadIdx.x;   // 16384 threads, coalesced
    float h = h0[ch];
    #pragma unroll 4
    for (int c = 0; c < NCHUNK; ++c) {
        const size_t o = (size_t)c * CH + ch;
        const float a = Aarr[o];
        const float b = Barr[o];
        Barr[o] = h;           // start value for chunk c
        h = a * h + b;         // compose
    }
}

// ---------------- Phase 3: replay each chunk from its known start, write all h_t ----------------
__global__ __launch_bounds__(BLOCK) void fm_phase3(const float* __restrict__ f,
                                                   const float* __restrict__ x,
                                                   const float* __restrict__ Hstart,
                                                   float* __restrict__ out) {
    __shared__ float4 bF[DSLOT][BLOCK];
    __shared__ float4 bX[DSLOT][BLOCK];
    const int tid = threadIdx.x;
    // Reverse chunk order vs phase 1: phase 1 streamed f/x in increasing-chunk order,
    // so the highest chunks are the ones still resident in the 192MB L2 -> read them first.
    const int chunk  = (NCHUNK - 1) - blockIdx.y;
    const int chBase = blockIdx.x * CHW + tid * CPT;

    const float* fp = f   + (size_t)chunk * TCHUNK * CH + chBase;
    const float* xp = x   + (size_t)chunk * TCHUNK * CH + chBase;
    float*       op = out + (size_t)chunk * TCHUNK * CH + chBase;

    const uint32_t lF = (uint32_t)(uintptr_t)&bF[0][tid];
    const uint32_t lX = (uint32_t)(uintptr_t)&bX[0][tid];
    const uint32_t slotBytes = (uint32_t)(sizeof(float4) * BLOCK);

    #pragma unroll
    for (int p = 0; p < KAHEAD; ++p) {
        async_ld16(lF + (uint32_t)p * slotBytes, fp + (size_t)p * CH);
        async_ld16(lX + (uint32_t)p * slotBytes, xp + (size_t)p * CH);
    }

    float4 h4 = *(const float4*)(Hstart + (size_t)chunk * CH + chBase);

    for (int i = 0; i < TCHUNK; ++i) {
        if (i < TCHUNK - KAHEAD)       wait_async_steady();
        else if (i == TCHUNK - KAHEAD) wait_async_all();

        const int slot = i & (DSLOT - 1);
        const float4 f4 = bF[slot][tid];
        const float4 x4 = bX[slot][tid];

        // h <- h + f*(x - h)  (one sub + one FMA per channel)
        h4.x += f4.x * (x4.x - h4.x);
        h4.y += f4.y * (x4.y - h4.y);
        h4.z += f4.z * (x4.z - h4.z);
        h4.w += f4.w * (x4.w - h4.w);

        // Output is written once and never re-read: non-temporal B128 store so the
        // 256MB result stream doesn't evict the f/x tail we want to hit in L2.
        // (native ext_vector type: __builtin_nontemporal_store rejects HIP_vector_type)
        v4f hv; hv.x = h4.x; hv.y = h4.y; hv.z = h4.z; hv.w = h4.w;
        __builtin_nontemporal_store(hv, (v4f*)(op + (size_t)i * CH));

        const int nr = i + KAHEAD;
        if (nr < TCHUNK) {
            wait_ds_all();
            const int ns = nr & (DSLOT - 1);
            async_ld16(lF + (uint32_t)ns * slotBytes, fp + (size_t)nr * CH);
            async_ld16(lX + (uint32_t)ns * slotBytes, xp + (size_t)nr * CH);
        }
    }
}

extern "C" void kernel_launch(void* const* d_in, const int* in_sizes, int n_in,
                              void* d_out, int out_size, void* d_ws, size_t ws_size,
                              hipStream_t stream) {
    const float* f  = (const float*)d_in[0];
    const float* x  = (const float*)d_in[1];
    const float* h0 = (const float*)d_in[2];
    float* out = (float*)d_out;

    // Workspace: A (32*16384 fp32 = 2MB) then B (2MB); fully rewritten each call.
    float* Aarr = (float*)d_ws;
    float* Barr = Aarr + (size_t)NCHUNK * CH;

    dim3 grid(NCB, NCHUNK);          // 16 x 32 = 512 workgroups, 8 waves each
    fm_phase1<<<grid, BLOCK, 0, stream>>>(f, x, Aarr, Barr);
    fm_phase2<<<CH / BLOCK, BLOCK, 0, stream>>>(Aarr, Barr, h0);
    fm_phase3<<<grid, BLOCK, 0, stream>>>(f, x, Barr, out);
}